// GraphAttentionNetwork_81965155877402
// MI455X (gfx1250) — compile-verified
//
#include <hip/hip_runtime.h>
#include <hip/hip_bf16.h>

// ---------------------------------------------------------------------------
// GAT on gfx1250 (MI455X, wave32).
//  - GEMMs use V_WMMA_F32_16X16X4_F32 (f32 in/out, matches reference math)
//  - Edge phase: 1 wave == 1 edge, lane == feature-in-head (UNITS == 32)
// ---------------------------------------------------------------------------

typedef __attribute__((ext_vector_type(2))) float v2f;
typedef __attribute__((ext_vector_type(8))) float v8f;

#define GAT_N     50000
#define GAT_FIN   256
#define GAT_E     1000000
#define GAT_UNITS 32
#define GAT_HEADS 8
#define GAT_D     256      // UNITS*HEADS
#define GAT_OUT   64
#define GAT_K     256      // contraction dim for every GEMM here

// ---------------------------------------------------------------------------
// Generic  C[M x Nout] = act(A[M x 256] @ B[256 x Nout] + bias)
// 4 waves / block, each wave owns one 16x16 output tile.
// B panel (256 x 16 = 16 KB) staged in LDS, shared by the 4 waves.
// ---------------------------------------------------------------------------
__global__ __launch_bounds__(128) void wmma_gemm_kernel(
    const float* __restrict__ A, const float* __restrict__ B,
    const float* __restrict__ bias, float* __restrict__ C,
    int M, int Nout, int relu)
{
    __shared__ float Bs[GAT_K * 16];

    const int n0 = blockIdx.x * 16;

    // cooperative stage of the B panel (columns n0..n0+15, all 256 rows)
    for (int i = threadIdx.x; i < GAT_K * 16; i += 128) {
        int k = i >> 4;
        int n = i & 15;
        Bs[i] = B[(size_t)k * Nout + n0 + n];
    }
    __syncthreads();

    const int wave  = threadIdx.x >> 5;
    const int lane  = threadIdx.x & 31;
    const int laneM = lane & 15;     // row (A) / col (B) inside tile
    const int laneH = lane >> 4;     // K-halve select (0/1)

    const int m0 = (blockIdx.y * 4 + wave) * 16;
    if (m0 >= M) return;             // wave-uniform guard (EXEC stays all-1)

    const float* Arow = A + (size_t)(m0 + laneM) * GAT_K + 2 * laneH;

    v8f c = {};
#pragma unroll
    for (int k0 = 0; k0 < GAT_K; k0 += 4) {
        if ((k0 & 63) == 0) __builtin_prefetch(Arow + k0 + 64, 0, 0);

        // A 16x4 tile: lane (M=laneM), components K = k0 + 2*laneH + {0,1}
        const float2 av = *(const float2*)(Arow + k0);
        v2f a; a.x = av.x; a.y = av.y;

        // B 4x16 tile from LDS: lane (N=laneM), rows K = k0 + 2*laneH + {0,1}
        v2f b;
        b.x = Bs[(k0 + 2 * laneH + 0) * 16 + laneM];
        b.y = Bs[(k0 + 2 * laneH + 1) * 16 + laneM];

        c = __builtin_amdgcn_wmma_f32_16x16x4_f32(false, a, false, b,
                                                  (short)0, c, false, false);
    }

    // C/D layout: VGPR r -> lanes 0-15: M=r, lanes 16-31: M=r+8; N = laneM
    const int col = n0 + laneM;
#pragma unroll
    for (int r = 0; r < 8; ++r) {
        int row = m0 + r + 8 * laneH;
        float v = c[r];
        if (bias) v += bias[col];
        if (relu) v = fmaxf(v, 0.0f);
        C[(size_t)row * Nout + col] = v;
    }
}

// ---------------------------------------------------------------------------
// Repack Wk[l] (HEADS, 256, 32) -> row-major B (256 x 256), n = h*32+u
// ---------------------------------------------------------------------------
__global__ __launch_bounds__(256) void pack_wk_kernel(
    const float* __restrict__ Wk_l, float* __restrict__ Bp)
{
    int i = blockIdx.x * blockDim.x + threadIdx.x;
    if (i >= GAT_D * GAT_D) return;
    int k = i >> 8;
    int n = i & 255;
    int h = n >> 5;
    int u = n & 31;
    Bp[i] = Wk_l[((size_t)h * GAT_D + k) * GAT_UNITS + u];
}

__global__ void zero_kernel(float* __restrict__ p, size_t n)
{
    size_t i = (size_t)blockIdx.x * blockDim.x + threadIdx.x;
    size_t stride = (size_t)gridDim.x * blockDim.x;
    for (; i < n; i += stride) p[i] = 0.0f;
}

// ---------------------------------------------------------------------------
// Edge pass 1: attention scores + segment-sum of exp-scores over src.
// One wave per edge; lane u handles feature u of each head.
//   s = exp(clip(leaky_relu(ew * (xt[src].a[:32] + xt[dst].a[32:]), 0.2), ±2))
// ---------------------------------------------------------------------------
__global__ __launch_bounds__(256) void edge_attn_kernel(
    const float* __restrict__ xt, const int* __restrict__ edges,
    const float* __restrict__ ew, const float* __restrict__ Wa_l,
    float* __restrict__ sbuf, float* __restrict__ ssum)
{
    int wid  = (blockIdx.x * blockDim.x + threadIdx.x) >> 5;
    int lane = threadIdx.x & 31;
    if (wid >= GAT_E) return;

    const int   src = edges[2 * wid + 0];
    const int   dst = edges[2 * wid + 1];
    const float w   = ew[wid];
    const float* xs = xt + (size_t)src * GAT_D;
    const float* xd = xt + (size_t)dst * GAT_D;

#pragma unroll
    for (int h = 0; h < GAT_HEADS; ++h) {
        float a0 = Wa_l[h * 64 + lane];        // a[0:32]
        float a1 = Wa_l[h * 64 + 32 + lane];   // a[32:64]
        float v  = (xs[h * 32 + lane] * a0 + xd[h * 32 + lane] * a1) * w;

        // wave32 butterfly reduction
        for (int off = 16; off > 0; off >>= 1)
            v += __shfl_xor(v, off, 32);

        float sv = v > 0.0f ? v : 0.2f * v;            // leaky_relu(0.2)
        sv = __expf(fminf(fmaxf(sv, -2.0f), 2.0f));    // exp(clip)

        if (lane == 0) {
            sbuf[(size_t)wid * GAT_HEADS + h] = sv;
            atomicAdd(&ssum[(size_t)src * GAT_HEADS + h], sv);
        }
    }
}

// ---------------------------------------------------------------------------
// Edge pass 2: acc[src] += xt[dst] * (s / ssum[src])  per head
// ---------------------------------------------------------------------------
__global__ __launch_bounds__(256) void edge_aggr_kernel(
    const float* __restrict__ xt, const int* __restrict__ edges,
    const float* __restrict__ sbuf, const float* __restrict__ ssum,
    float* __restrict__ acc)
{
    int wid  = (blockIdx.x * blockDim.x + threadIdx.x) >> 5;
    int lane = threadIdx.x & 31;
    if (wid >= GAT_E) return;

    const int src = edges[2 * wid + 0];
    const int dst = edges[2 * wid + 1];
    const float* xd = xt  + (size_t)dst * GAT_D;
    float*       as = acc + (size_t)src * GAT_D;

#pragma unroll
    for (int h = 0; h < GAT_HEADS; ++h) {
        float norm = sbuf[(size_t)wid * GAT_HEADS + h] /
                     ssum[(size_t)src * GAT_HEADS + h];
        atomicAdd(&as[h * 32 + lane], xd[h * 32 + lane] * norm);
    }
}

// x = relu(acc) + x
__global__ __launch_bounds__(256) void combine_kernel(
    float* __restrict__ x, const float* __restrict__ acc, size_t n)
{
    size_t i = (size_t)blockIdx.x * blockDim.x + threadIdx.x;
    if (i < n) x[i] = fmaxf(acc[i], 0.0f) + x[i];
}

// ---------------------------------------------------------------------------
extern "C" void kernel_launch(void* const* d_in, const int* in_sizes, int n_in,
                              void* d_out, int out_size, void* d_ws, size_t ws_size,
                              hipStream_t stream)
{
    (void)in_sizes; (void)n_in; (void)out_size; (void)ws_size;

    const float* node_states = (const float*)d_in[0];   // (1,N,256)
    const int*   edges       = (const int*)  d_in[1];   // (1,E,2)  [src,dst]
    const float* eweights    = (const float*)d_in[2];   // (1,E)
    /* d_in[3] = pos_cls: unused by the reference */
    const float* Wp = (const float*)d_in[4];            // (256,256)
    const float* bp = (const float*)d_in[5];            // (256)
    const float* Wk = (const float*)d_in[6];            // (2,8,256,32)
    const float* Wa = (const float*)d_in[7];            // (2,8,64,1)
    const float* Wo = (const float*)d_in[8];            // (256,64)
    const float* bo = (const float*)d_in[9];            // (64)
    float* out = (float*)d_out;                         // (1,N,64)

    // ---- workspace carve-up (~187 MB) ----
    char* ws = (char*)d_ws;
    float* x    = (float*)ws; ws += (size_t)GAT_N * GAT_D * 4;
    float* xt   = (float*)ws; ws += (size_t)GAT_N * GAT_D * 4;
    float* acc  = (float*)ws; ws += (size_t)GAT_N * GAT_D * 4;
    float* sbuf = (float*)ws; ws += (size_t)GAT_E * GAT_HEADS * 4;
    float* ssum = (float*)ws; ws += (size_t)GAT_N * GAT_HEADS * 4;
    float* Bp   = (float*)ws; ws += (size_t)GAT_D * GAT_D * 4;

    const int mTiles  = GAT_N / 16;                 // 3125
    const dim3 gemmBlk(128);
    const dim3 gemmGridD(GAT_D / 16,  (mTiles + 3) / 4);
    const dim3 gemmGridO(GAT_OUT / 16,(mTiles + 3) / 4);
    const int edgeGrid = (GAT_E * 32 + 255) / 256;  // 1 wave per edge, 8 waves/blk

    // 1) projection: x = relu(x0 @ Wp + bp)
    wmma_gemm_kernel<<<gemmGridD, gemmBlk, 0, stream>>>(
        node_states, Wp, bp, x, GAT_N, GAT_D, 1);

    // 2) GAT layers
    for (int l = 0; l < 2; ++l) {
        pack_wk_kernel<<<(GAT_D * GAT_D + 255) / 256, 256, 0, stream>>>(
            Wk + (size_t)l * GAT_HEADS * GAT_D * GAT_UNITS, Bp);

        wmma_gemm_kernel<<<gemmGridD, gemmBlk, 0, stream>>>(
            x, Bp, nullptr, xt, GAT_N, GAT_D, 0);

        zero_kernel<<<2048, 256, 0, stream>>>(acc,  (size_t)GAT_N * GAT_D);
        zero_kernel<<<512,  256, 0, stream>>>(ssum, (size_t)GAT_N * GAT_HEADS);

        const float* Wa_l = Wa + (size_t)l * GAT_HEADS * 2 * GAT_UNITS;
        edge_attn_kernel<<<edgeGrid, 256, 0, stream>>>(
            xt, edges, eweights, Wa_l, sbuf, ssum);
        edge_aggr_kernel<<<edgeGrid, 256, 0, stream>>>(
            xt, edges, sbuf, ssum, acc);

        combine_kernel<<<(GAT_N * GAT_D + 255) / 256, 256, 0, stream>>>(
            x, acc, (size_t)GAT_N * GAT_D);
    }

    // 3) output head: out = x @ Wo + bo
    wmma_gemm_kernel<<<gemmGridO, gemmBlk, 0, stream>>>(
        x, Wo, bo, out, GAT_N, GAT_OUT, 0);
}